// ResidualVectorQuantizer_30116310679558
// MI455X (gfx1250) — compile-verified
//
#include <hip/hip_runtime.h>

typedef float v2f __attribute__((ext_vector_type(2)));
typedef float v8f __attribute__((ext_vector_type(8)));

#define BB 32
#define TT 2048
#define DD 256
#define KK 1024
#define NCB 8
#define NROWS (BB * TT)              // 65536
#define ROWS_PER_WG 128              // 8 waves * 16 rows
#define ARG_GRID (NROWS / ROWS_PER_WG)   // 512
#define UPD_ELEMS 8                  // floats per thread in update kernel
#define UPD_GRID ((size_t)NROWS * DD / (256 * UPD_ELEMS))  // 8192
#define LDSB_STRIDE (DD + 2)         // pad to dodge bank conflicts

// ---------------------------------------------------------------------------
// init: residual = z, quantized = 0
// ---------------------------------------------------------------------------
__global__ __launch_bounds__(256) void rvq_init_kernel(
    const float* __restrict__ z, float* __restrict__ res,
    float* __restrict__ quant) {
  size_t i = ((size_t)blockIdx.x * 256 + threadIdx.x) * 4;
  float4 v = *(const float4*)(z + i);
  *(float4*)(res + i) = v;
  float4 zf; zf.x = 0.f; zf.y = 0.f; zf.z = 0.f; zf.w = 0.f;
  *(float4*)(quant + i) = zf;
}

// ---------------------------------------------------------------------------
// wnorm: ||W_k||^2 for all NC*K codewords
// ---------------------------------------------------------------------------
__global__ __launch_bounds__(256) void rvq_wnorm_kernel(
    const float* __restrict__ codebooks, float* __restrict__ wnorm) {
  int k = blockIdx.x * 256 + threadIdx.x;  // 0..8191
  const float* w = codebooks + (size_t)k * DD;
  float s = 0.f;
#pragma unroll 8
  for (int i = 0; i < DD / 4; ++i) {
    float4 v = *(const float4*)(w + 4 * i);
    s += v.x * v.x + v.y * v.y + v.z * v.z + v.w * v.w;
  }
  wnorm[k] = s;
}

// ---------------------------------------------------------------------------
// Fused GEMM + argmin for one codebook, f32 WMMA 16x16x4.
// Each wave: one 16-row tile, A fragment held in 64 v2f registers.
// Each WG: stages 16-codeword B tiles in LDS, shared by 8 waves.
// ---------------------------------------------------------------------------
__global__ __launch_bounds__(256) void rvq_argmin_kernel(
    const float* __restrict__ res,   // [N, D]
    const float* __restrict__ W,     // [K, D] (codebook c)
    const float* __restrict__ wnorm, // [K]    (codebook c)
    int* __restrict__ idx_out,       // [N]
    float* __restrict__ idx_out_f,   // d_out index region, row stride NCB
    int cbook) {
  __shared__ float ldsB[16 * LDSB_STRIDE];

  const int tid  = threadIdx.x;
  const int lane = tid & 31;
  const int wave = tid >> 5;                 // 0..7
  const int lm   = lane & 15;                // col / row-in-tile selector
  const int lh   = lane >> 4;                // K-pair selector
  const int rowBase = blockIdx.x * ROWS_PER_WG + wave * 16;

  // Load this wave's A fragments once: res[rowBase+lm][4d + 2lh + {0,1}]
  v2f aF[64];
  {
    const float* arow = res + (size_t)(rowBase + lm) * DD + 2 * lh;
#pragma unroll
    for (int d = 0; d < 64; ++d) aF[d] = *(const v2f*)(arow + 4 * d);
  }

  float bestV[8];
  int   bestI[8];
#pragma unroll
  for (int j = 0; j < 8; ++j) { bestV[j] = 3.4e38f; bestI[j] = 0; }

  for (int n0 = 0; n0 < KK; n0 += 16) {
    __syncthreads();
    // Cooperative B-tile load: 16 codewords x 256 floats (as float4s)
    for (int i = tid; i < 16 * (DD / 4); i += 256) {
      int r = i >> 6, c4 = i & 63;
      float4 v = *(const float4*)(W + (size_t)(n0 + r) * DD + c4 * 4);
      float* dst = &ldsB[r * LDSB_STRIDE + c4 * 4];
      dst[0] = v.x; dst[1] = v.y; dst[2] = v.z; dst[3] = v.w;
    }
    __syncthreads();

    const float wn = wnorm[n0 + lm];
    v8f acc = {0.f, 0.f, 0.f, 0.f, 0.f, 0.f, 0.f, 0.f};
    const float* brow = &ldsB[lm * LDSB_STRIDE + 2 * lh];
#pragma unroll
    for (int d = 0; d < 64; ++d) {
      v2f b; b.x = brow[4 * d]; b.y = brow[4 * d + 1];
      acc = __builtin_amdgcn_wmma_f32_16x16x4_f32(
          false, aF[d], false, b, (short)0, acc, false, false);
    }

    // dist(row = rowBase + j + 8*lh, col = n0 + lm) = ||w||^2 - 2 dot
#pragma unroll
    for (int j = 0; j < 8; ++j) {
      float dist = wn - 2.0f * acc[j];
      if (dist < bestV[j]) { bestV[j] = dist; bestI[j] = n0 + lm; }
    }
  }

  // Butterfly min-reduce over the 16-lane column group, first-index tiebreak
#pragma unroll
  for (int j = 0; j < 8; ++j) {
    float v = bestV[j];
    int   bi = bestI[j];
#pragma unroll
    for (int m = 1; m <= 8; m <<= 1) {
      float ov = __shfl_xor(v, m, 32);
      int   oi = __shfl_xor(bi, m, 32);
      if (ov < v || (ov == v && oi < bi)) { v = ov; bi = oi; }
    }
    if (lm == 0) {
      int row = rowBase + j + 8 * lh;
      idx_out[row] = bi;
      idx_out_f[(size_t)row * NCB + cbook] = (float)bi;
    }
  }
}

// ---------------------------------------------------------------------------
// Residual/quantized update + per-WG sum of squares (deterministic)
// ---------------------------------------------------------------------------
__global__ __launch_bounds__(256) void rvq_update_kernel(
    float* __restrict__ res, float* __restrict__ quant,
    const float* __restrict__ W, const int* __restrict__ idxbuf,
    float* __restrict__ partials) {
  __shared__ float red[256];
  const int tid = threadIdx.x;
  size_t base = ((size_t)blockIdx.x * 256 + tid) * UPD_ELEMS;
  int n = (int)(base >> 8);       // row (D = 256)
  int d = (int)(base & 255);
  int idx = idxbuf[n];
  const float* wp = W + (size_t)idx * DD + d;

  float s = 0.f;
#pragma unroll
  for (int h = 0; h < UPD_ELEMS / 4; ++h) {
    float4 r = *(float4*)(res + base + 4 * h);
    float4 w = *(const float4*)(wp + 4 * h);
    float4 q = *(float4*)(quant + base + 4 * h);
    r.x -= w.x; r.y -= w.y; r.z -= w.z; r.w -= w.w;
    q.x += w.x; q.y += w.y; q.z += w.z; q.w += w.w;
    *(float4*)(res + base + 4 * h) = r;
    *(float4*)(quant + base + 4 * h) = q;
    s += r.x * r.x + r.y * r.y + r.z * r.z + r.w * r.w;
  }
  red[tid] = s;
  __syncthreads();
  for (int off = 128; off > 0; off >>= 1) {
    if (tid < off) red[tid] += red[tid + off];
    __syncthreads();
  }
  if (tid == 0) partials[blockIdx.x] = red[0];
}

// ---------------------------------------------------------------------------
// Final loss: sum partials per codebook in fixed order, loss = sum of means
// ---------------------------------------------------------------------------
__global__ __launch_bounds__(256) void rvq_loss_kernel(
    const float* __restrict__ partials, float* __restrict__ loss_out) {
  __shared__ float red[256];
  __shared__ float total;
  const int tid = threadIdx.x;
  if (tid == 0) total = 0.f;
  __syncthreads();
  for (int c = 0; c < NCB; ++c) {
    float s = 0.f;
    for (int i = tid; i < (int)UPD_GRID; i += 256) s += partials[c * UPD_GRID + i];
    red[tid] = s;
    __syncthreads();
    for (int off = 128; off > 0; off >>= 1) {
      if (tid < off) red[tid] += red[tid + off];
      __syncthreads();
    }
    if (tid == 0) total += red[0] * (1.0f / ((float)NROWS * (float)DD));
    __syncthreads();
  }
  if (tid == 0) loss_out[0] = total;
}

// ---------------------------------------------------------------------------
extern "C" void kernel_launch(void* const* d_in, const int* in_sizes, int n_in,
                              void* d_out, int out_size, void* d_ws,
                              size_t ws_size, hipStream_t stream) {
  (void)in_sizes; (void)n_in; (void)out_size; (void)ws_size;
  const float* z         = (const float*)d_in[0];   // [B,T,D]
  const float* codebooks = (const float*)d_in[1];   // [NC,K,D]

  float* out   = (float*)d_out;
  float* quant = out;                                       // N*D
  float* idxf  = out + (size_t)NROWS * DD;                  // N*NC
  float* lossp = idxf + (size_t)NROWS * NCB;                // 1

  char* ws = (char*)d_ws;
  float* res      = (float*)ws;                ws += (size_t)NROWS * DD * 4;
  int*   idxbuf   = (int*)ws;                  ws += (size_t)NROWS * 4;
  float* wnorm    = (float*)ws;                ws += (size_t)NCB * KK * 4;
  float* partials = (float*)ws;                // NCB * UPD_GRID floats

  // residual = z, quantized = 0
  rvq_init_kernel<<<(NROWS * DD) / 1024, 256, 0, stream>>>(z, res, quant);
  // codeword norms
  rvq_wnorm_kernel<<<(NCB * KK) / 256, 256, 0, stream>>>(codebooks, wnorm);

  for (int c = 0; c < NCB; ++c) {
    const float* W  = codebooks + (size_t)c * KK * DD;
    const float* wn = wnorm + (size_t)c * KK;
    rvq_argmin_kernel<<<ARG_GRID, 256, 0, stream>>>(res, W, wn, idxbuf, idxf, c);
    rvq_update_kernel<<<(int)UPD_GRID, 256, 0, stream>>>(
        res, quant, W, idxbuf, partials + (size_t)c * UPD_GRID);
  }
  rvq_loss_kernel<<<1, 256, 0, stream>>>(partials, lossp);
}